// Chamfer_Distance_78297253806557
// MI455X (gfx1250) — compile-verified
//
#include <hip/hip_runtime.h>

// CDNA5 / gfx1250 Chamfer distance, single pass per direction.
// d2(p,t) tile (16x16) = one V_WMMA_F32_16X16X4_F32 with augmented operands:
//   A row m = [px, py, |p|^2, 1]        (16x4, 2 VGPRs: lanes 0-15 K=0,1 ; lanes 16-31 K=2,3)
//   B col n = [-2tx, -2ty, 1, |t|^2]^T  (4x16, 2 VGPRs, same half split)
// B operand is pre-augmented into LDS once per workgroup -> inner loop is
// ds_load_b64 + 2x wmma + mins (two A row-tiles per wave share each B load).

typedef __attribute__((ext_vector_type(2))) float v2f;
typedef __attribute__((ext_vector_type(8))) float v8f;

#define BATCH 16
#define NPTS  4096
#define CTILES (NPTS / 16)          // 256 column tiles
#define WAVES_PER_BLOCK 4
#define TILES_PER_WAVE 2
#define ROWTILES_PER_BLOCK (WAVES_PER_BLOCK * TILES_PER_WAVE)   // 8

__global__ void chamfer_init_out(float* out) {
    out[0] = 0.0f;
    out[1] = 0.0f;
}

__global__ __launch_bounds__(32 * WAVES_PER_BLOCK)
void chamfer_kernel(const float2* __restrict__ pred,
                    const float2* __restrict__ target,
                    float* __restrict__ out)
{
    // Augmented B operand, interleaved [n][half] so lane l / half h reads
    // byte l*16 + h*8 within a tile: conflict-free across all 64 LDS banks.
    __shared__ float2 lb[NPTS * 2];   // 64 KB

    const int dir = blockIdx.z;                 // 0: pred->target, 1: target->pred
    const float2* rows = dir == 0 ? pred : target;
    const float2* cols = dir == 0 ? target : pred;
    const int b = blockIdx.y;

    const int tid  = threadIdx.x;
    const int lane = tid & 31;
    const int wave = tid >> 5;
    const int half = lane >> 4;                 // 0: lanes 0-15, 1: lanes 16-31
    const int l    = lane & 15;

    // ---- Stage augmented B operand into LDS (once per workgroup) ----
    const float2* cb = cols + (size_t)b * NPTS;
    for (int n = tid; n < NPTS; n += 32 * WAVES_PER_BLOCK) {
        const float2 t = cb[n];
        lb[n * 2 + 0] = make_float2(-2.0f * t.x, -2.0f * t.y);        // K0, K1
        lb[n * 2 + 1] = make_float2(1.0f, t.x * t.x + t.y * t.y);     // K2, K3
    }
    __syncthreads();

    // ---- Two A row-tiles per wave (held in registers for the whole loop) ----
    const int rt0 = (blockIdx.x * WAVES_PER_BLOCK + wave) * TILES_PER_WAVE;
    const float2* rb = rows + (size_t)b * NPTS;
    v2f a0, a1;
    {
        const float2 p0 = rb[(rt0 + 0) * 16 + l];
        const float2 p1 = rb[(rt0 + 1) * 16 + l];
        if (half == 0) {
            a0.x = p0.x; a0.y = p0.y;
            a1.x = p1.x; a1.y = p1.y;
        } else {
            a0.x = p0.x * p0.x + p0.y * p0.y; a0.y = 1.0f;
            a1.x = p1.x * p1.x + p1.y * p1.y; a1.y = 1.0f;
        }
    }

    v8f rmin0, rmin1;
#pragma unroll
    for (int k = 0; k < 8; ++k) { rmin0[k] = 3.0e38f; rmin1[k] = 3.0e38f; }

    // ---- Sweep all column tiles: 1 LDS load, 2 WMMAs, running mins ----
    for (int ct = 0; ct < CTILES; ++ct) {
        const float2 t = lb[(ct * 16 + l) * 2 + half];
        v2f bb; bb.x = t.x; bb.y = t.y;

        v8f c = {};
        const v8f d0 = __builtin_amdgcn_wmma_f32_16x16x4_f32(
            false, a0, false, bb, (short)0, c, false, false);
        const v8f d1 = __builtin_amdgcn_wmma_f32_16x16x4_f32(
            false, a1, false, bb, (short)0, c, false, false);

#pragma unroll
        for (int k = 0; k < 8; ++k) {
            rmin0[k] = fminf(rmin0[k], d0[k]);
            rmin1[k] = fminf(rmin1[k], d1[k]);
        }
    }

    // ---- Row-min: reduce over the 16 lanes of each half (N dimension) ----
#pragma unroll
    for (int off = 8; off >= 1; off >>= 1) {
#pragma unroll
        for (int k = 0; k < 8; ++k) {
            rmin0[k] = fminf(rmin0[k], __shfl_xor(rmin0[k], off, 16));
            rmin1[k] = fminf(rmin1[k], __shfl_xor(rmin1[k], off, 16));
        }
    }

    // Sum the 8 row-mins of this half for both tiles, combine halves, one atomic.
    float s = 0.0f;
#pragma unroll
    for (int k = 0; k < 8; ++k) s += rmin0[k] + rmin1[k];
    s += __shfl_xor(s, 16, 32);

    if (lane == 0) {
        const float scale = 1.0f / (float)(NPTS * BATCH);
        atomicAdd(out + dir, s * scale);
    }
}

extern "C" void kernel_launch(void* const* d_in, const int* in_sizes, int n_in,
                              void* d_out, int out_size, void* d_ws, size_t ws_size,
                              hipStream_t stream) {
    (void)in_sizes; (void)n_in; (void)out_size; (void)d_ws; (void)ws_size;

    const float2* pred   = (const float2*)d_in[0];
    const float2* target = (const float2*)d_in[1];
    float* out = (float*)d_out;

    chamfer_init_out<<<1, 1, 0, stream>>>(out);

    dim3 grid(CTILES / ROWTILES_PER_BLOCK, BATCH, 2);   // (32, 16, 2)
    dim3 block(32 * WAVES_PER_BLOCK);                   // 128 threads = 4 waves

    chamfer_kernel<<<grid, block, 0, stream>>>(pred, target, out);
}